// MemoryBank_8048768713555
// MI455X (gfx1250) — compile-verified
//
#include <hip/hip_runtime.h>
#include <hip/hip_bf16.h>
#include <math.h>

#define T_TOKENS   8192
#define HIDDEN_DIM 4096
#define NUM_SLOTS  128
#define K_RET      4
#define TOPK       2048
#define EMA_ALPHA  0.1f
#define RANK_CHUNKS 8
#define SUMS_ELEMS (NUM_SLOTS * HIDDEN_DIM)   // 524288 floats = 2MB

typedef __attribute__((ext_vector_type(2))) float v2f;
typedef __attribute__((ext_vector_type(8))) float v8f;

// ---------------------------------------------------------------------------
// Kernel 1: importance[t] = ||h_t|| * (1 + entropy/log4) + sigmoid(h_t.W + b)
// One wave32 per token; float4 streaming loads; shfl_xor reduction.
// ---------------------------------------------------------------------------
__global__ void imp_kernel(const float* __restrict__ h,
                           const float* __restrict__ attn,
                           const float* __restrict__ Wimp,
                           const float* __restrict__ bimp,
                           float* __restrict__ imp) {
    int gtid = blockIdx.x * blockDim.x + threadIdx.x;
    int tok  = gtid >> 5;
    int lane = threadIdx.x & 31;
    if (tok >= T_TOKENS) return;

    const float* hrow = h + (size_t)tok * HIDDEN_DIM;
    float hh = 0.0f, hw = 0.0f;
    #pragma unroll 4
    for (int i = lane * 4; i < HIDDEN_DIM; i += 128) {
        float4 v = *(const float4*)(hrow + i);
        float4 w = *(const float4*)(Wimp + i);
        hh += v.x * v.x + v.y * v.y + v.z * v.z + v.w * v.w;
        hw += v.x * w.x + v.y * w.y + v.z * w.z + v.w * w.w;
    }
    #pragma unroll
    for (int off = 16; off > 0; off >>= 1) {
        hh += __shfl_xor(hh, off);
        hw += __shfl_xor(hw, off);
    }
    if (lane == 0) {
        float4 a = *(const float4*)(attn + (size_t)tok * K_RET);
        float ent = -(a.x * __logf(a.x + 1e-8f) + a.y * __logf(a.y + 1e-8f) +
                      a.z * __logf(a.z + 1e-8f) + a.w * __logf(a.w + 1e-8f));
        float surprise = ent / 1.3862943611198906f;   // log(4)
        float mag   = sqrtf(hh);
        float score = hw + bimp[0];
        float sig   = 1.0f / (1.0f + __expf(-score));
        imp[tok] = mag * (1.0f + surprise) + sig;
    }
}

// ---------------------------------------------------------------------------
// Kernel 2a: partial rank over one chunk of the comparison range.
// rank contribution of u in [chunk*1024, chunk*1024+1024) for token t.
// grid: (T/256, RANK_CHUNKS). Deterministic, no atomics.
// ---------------------------------------------------------------------------
__global__ void prank_kernel(const float* __restrict__ imp,
                             int* __restrict__ prank) {
    __shared__ float simp[T_TOKENS / RANK_CHUNKS];   // 1024 floats
    int t     = blockIdx.x * blockDim.x + threadIdx.x;
    int chunk = blockIdx.y;
    int base  = chunk * (T_TOKENS / RANK_CHUNKS);
    float mine = imp[t];

    for (int i = threadIdx.x; i < T_TOKENS / RANK_CHUNKS; i += blockDim.x)
        simp[i] = imp[base + i];
    __syncthreads();

    int rank = 0;
    #pragma unroll 8
    for (int j = 0; j < T_TOKENS / RANK_CHUNKS; ++j) {
        float v = simp[j];
        int   u = base + j;
        rank += ((v > mine) || (v == mine && u < t)) ? 1 : 0;
    }
    prank[chunk * T_TOKENS + t] = rank;
}

// ---------------------------------------------------------------------------
// Kernel 2b: combine partial ranks; scatter token to its rank slot.
// ---------------------------------------------------------------------------
__global__ void select_kernel(const int* __restrict__ prank,
                              int* __restrict__ sel) {
    int t = blockIdx.x * blockDim.x + threadIdx.x;
    int rank = 0;
    #pragma unroll
    for (int c = 0; c < RANK_CHUNKS; ++c)
        rank += prank[c * T_TOKENS + t];
    if (rank < TOPK) sel[rank] = t;
}

// ---------------------------------------------------------------------------
// Kernel 3: counts[n] = #{selected t : n in slot_indices[t]}. 128 threads.
// ---------------------------------------------------------------------------
__global__ void counts_kernel(const int* __restrict__ sel,
                              const int* __restrict__ slot_idx,
                              float* __restrict__ counts) {
    int n = threadIdx.x;
    float c = 0.0f;
    for (int i = 0; i < TOPK; ++i) {
        int t = sel[i];
        int4 s = *(const int4*)(slot_idx + (size_t)t * K_RET);
        c += (s.x == n || s.y == n || s.z == n || s.w == n) ? 1.0f : 0.0f;
    }
    counts[n] = c;
}

// ---------------------------------------------------------------------------
// Kernel 4: partial sums[n,d] over a K-chunk of selected tokens via
// V_WMMA_F32_16X16X4_F32 (M=16 slots, N=16 dims, K=4 tokens, exact fp32).
// grid: (HIDDEN_DIM/16, nchunks); one wave per (column tile, K-chunk).
// Membership synthesized as an 8-bit per-lane tile mask:
//   index v hits this lane's column iff (v&15)==(lane&15); tile bit = v>>4.
// A layout (ISA 7.12.2, 32-bit A 16x4): VGPR0 = K0 (lanes 0-15) / K2
// (lanes 16-31); VGPR1 = K1 / K3.  B symmetric.  C/D: VGPR r holds
// M = r + 8*(lane>=16), N = lane&15.
// ---------------------------------------------------------------------------
__global__ void wmma_sums_kernel(const float* __restrict__ h,
                                 const int* __restrict__ slot_idx,
                                 const int* __restrict__ sel,
                                 float* __restrict__ sums_base,
                                 int kk_per_chunk) {
    const int j     = blockIdx.x;            // dim tile 0..255
    const int chunk = blockIdx.y;            // K chunk
    const int lane  = threadIdx.x;           // 32 threads = 1 wave
    const int c15   = lane & 15;
    const int col   = (j << 4) + c15;
    const int kx    = (lane >> 4) << 1;      // K index 0 (lanes 0-15) or 2
    float* __restrict__ sums = sums_base + (size_t)chunk * SUMS_ELEMS;

    v8f c[8];
    #pragma unroll
    for (int s = 0; s < 8; ++s) c[s] = (v8f){0,0,0,0,0,0,0,0};

    const int kk0 = chunk * kk_per_chunk;
    for (int kk = kk0; kk < kk0 + kk_per_chunk; ++kk) {
        int tx = sel[4 * kk + kx];
        int ty = sel[4 * kk + kx + 1];
        int4 sx = *(const int4*)(slot_idx + (size_t)tx * K_RET);
        int4 sy = *(const int4*)(slot_idx + (size_t)ty * K_RET);

        unsigned mx = 0u, my = 0u;
        mx |= ((sx.x & 15) == c15) ? (1u << (sx.x >> 4)) : 0u;
        mx |= ((sx.y & 15) == c15) ? (1u << (sx.y >> 4)) : 0u;
        mx |= ((sx.z & 15) == c15) ? (1u << (sx.z >> 4)) : 0u;
        mx |= ((sx.w & 15) == c15) ? (1u << (sx.w >> 4)) : 0u;
        my |= ((sy.x & 15) == c15) ? (1u << (sy.x >> 4)) : 0u;
        my |= ((sy.y & 15) == c15) ? (1u << (sy.y >> 4)) : 0u;
        my |= ((sy.z & 15) == c15) ? (1u << (sy.z >> 4)) : 0u;
        my |= ((sy.w & 15) == c15) ? (1u << (sy.w >> 4)) : 0u;

        v2f b;
        b.x = h[(size_t)tx * HIDDEN_DIM + col];
        b.y = h[(size_t)ty * HIDDEN_DIM + col];

        #pragma unroll
        for (int s = 0; s < 8; ++s) {
            v2f a;
            a.x = (float)((mx >> s) & 1u);   // exact 0.0 / 1.0
            a.y = (float)((my >> s) & 1u);
            c[s] = __builtin_amdgcn_wmma_f32_16x16x4_f32(
                false, a, false, b, (short)0, c[s], false, false);
        }
    }

    const int mhalf = (lane >> 4) * 8;
    #pragma unroll
    for (int s = 0; s < 8; ++s) {
        #pragma unroll
        for (int r = 0; r < 8; ++r) {
            int slot = (s << 4) + mhalf + r;
            sums[(size_t)slot * HIDDEN_DIM + col] = c[s][r];
        }
    }
}

// ---------------------------------------------------------------------------
// Kernel 5: out = counts>0 ? 0.1*(sum_chunks partial)/max(counts,1) + 0.9*mem
//                          : mem
// ---------------------------------------------------------------------------
__global__ void finalize_kernel(const float* __restrict__ sums_base,
                                const float* __restrict__ counts,
                                const float* __restrict__ mem,
                                float* __restrict__ out,
                                int nchunks) {
    int i = blockIdx.x * blockDim.x + threadIdx.x;   // 0 .. 524287
    int slot = i >> 12;                              // / HIDDEN_DIM
    float acc = 0.0f;
    for (int cch = 0; cch < nchunks; ++cch)
        acc += sums_base[(size_t)cch * SUMS_ELEMS + i];
    float cnt = counts[slot];
    float cur = mem[i];
    float agg = acc / fmaxf(cnt, 1.0f);
    float upd = EMA_ALPHA * agg + (1.0f - EMA_ALPHA) * cur;
    out[i] = (cnt > 0.0f) ? upd : cur;
}

// ---------------------------------------------------------------------------
extern "C" void kernel_launch(void* const* d_in, const int* in_sizes, int n_in,
                              void* d_out, int out_size, void* d_ws, size_t ws_size,
                              hipStream_t stream) {
    const float* h     = (const float*)d_in[0];   // [8192, 4096]
    const float* attn  = (const float*)d_in[1];   // [8192, 4]
    const int*   sidx  = (const int*)  d_in[2];   // [8192, 4]
    const float* mem   = (const float*)d_in[3];   // [1, 128, 4096]
    const float* Wimp  = (const float*)d_in[4];   // [1, 4096]
    const float* bimp  = (const float*)d_in[5];   // [1]
    float* out = (float*)d_out;                   // [1, 128, 4096]

    // workspace layout
    char* ws = (char*)d_ws;
    float* imp    = (float*)(ws);                     // 8192 f32   (32KB)
    int*   prank  = (int*)  (ws + 32768);             // 8*8192 i32 (256KB)
    int*   sel    = (int*)  (ws + 294912);            // 2048 i32   (8KB)
    float* counts = (float*)(ws + 303104);            // 128 f32
    float* sums   = (float*)(ws + 524288);            // nchunks * 2MB partials

    // pick K-chunk count (power of 2, <=8) that fits in the workspace
    size_t avail = (ws_size > 524288) ? (ws_size - 524288) / (SUMS_ELEMS * 4) : 1;
    int nchunks = 1;
    while (nchunks < 8 && (size_t)(nchunks * 2) <= avail) nchunks *= 2;
    int kk_per_chunk = (TOPK / 4) / nchunks;

    // 1) importance: 8192 waves, 8 waves/block
    imp_kernel<<<T_TOKENS / 8, 256, 0, stream>>>(h, attn, Wimp, bimp, imp);
    // 2) top-K selection by exact rank (split + combine)
    prank_kernel<<<dim3(T_TOKENS / 256, RANK_CHUNKS), 256, 0, stream>>>(imp, prank);
    select_kernel<<<T_TOKENS / 256, 256, 0, stream>>>(prank, sel);
    // 3) slot membership counts
    counts_kernel<<<1, NUM_SLOTS, 0, stream>>>(sel, sidx, counts);
    // 4) fp32 WMMA segment-sum: 256 column tiles x nchunks K-chunks, 1 wave each
    wmma_sums_kernel<<<dim3(HIDDEN_DIM / 16, nchunks), 32, 0, stream>>>(
        h, sidx, sel, sums, kk_per_chunk);
    // 5) chunk reduction + EMA blend
    finalize_kernel<<<(NUM_SLOTS * HIDDEN_DIM) / 256, 256, 0, stream>>>(
        sums, counts, mem, out, nchunks);
}